// HyperLayer_22763326669372
// MI455X (gfx1250) — compile-verified
//
#include <hip/hip_runtime.h>
#include <stdint.h>

typedef float v2f __attribute__((ext_vector_type(2)));
typedef float v8f __attribute__((ext_vector_type(8)));

#define B_      16
#define N_      4096
#define CHUNK_  32
#define K_      128
#define NPTS_   20
#define L_      640
#define OUT_    1024
#define IN_     1024
#define RNGF    1024.0f
#define RNGM1   1023.0f
#define ONE_MEPS 0.999999f   // 1 - 1e-6

__global__ void init_out_kernel(const float* __restrict__ bias,
                                float* __restrict__ out) {
    int i = blockIdx.x * blockDim.x + threadIdx.x;
    if (i < B_ * OUT_) out[i] = bias[i & (OUT_ - 1)];
}

__launch_bounds__(256)
__global__ void hyper_kernel(const float* __restrict__ input,
                             const float* __restrict__ res,
                             const float* __restrict__ ug,
                             const float* __restrict__ ur,
                             float* __restrict__ out) {
    __shared__ float s_m[CHUNK_][2];
    __shared__ float sB[4][CHUNK_];      // WMMA B-matrix (4 x 32)
    __shared__ float s_t3[CHUNK_];
    __shared__ float s_value[CHUNK_];
    __shared__ float s_vr[CHUNK_];       // value / denom
    __shared__ float s_denom[CHUNK_];
    __shared__ int   s_x[L_];
    __shared__ int   s_y[L_];
    __shared__ unsigned long long s_h[L_];
    __shared__ float s_ds[L_];           // 0 if duplicate, else 1
    __shared__ float s_val[L_];

    const int tid = threadIdx.x;
    const int b   = blockIdx.x / K_;
    const int k   = blockIdx.x % K_;

    // ---- Phase 1: per-Gaussian params -> WMMA B-matrix columns ----
    if (tid < CHUNK_) {
        int c = tid;
        const float* rp = res + (((size_t)b * N_ + (size_t)k * CHUNK_ + c) * 4);
        float r0 = rp[0], r1 = rp[1], r2 = rp[2], r3 = rp[3];
        float m0 = RNGM1 / (1.f + __expf(-r0));     // sigmoid * (rng-1)
        float m1 = RNGM1 / (1.f + __expf(-r1));
        float z  = r2 + 2.0f;                        // SIGMA_BOOST
        float sp = (z > 20.f) ? z : log1pf(__expf(z));
        float sigma = (sp + 1e-6f) * RNGF;
        float isig  = 1.f / (1e-6f + sigma);         // same for both dims
        s_m[c][0] = m0;  s_m[c][1] = m1;
        sB[0][c] = isig;
        sB[1][c] = isig;
        sB[2][c] = -2.f * isig * m0;
        sB[3][c] = -2.f * isig * m1;
        s_t3[c]   = isig * (m0 * m0 + m1 * m1);
        s_value[c] = r3;
    }
    __syncthreads();

    // ---- Phase 2: generate the 640 integer points + 64-bit hashes ----
    for (int l = tid; l < L_; l += 256) {
        int c = l / NPTS_;
        int p = l % NPTS_;
        float m0 = s_m[c][0], m1 = s_m[c][1];
        float x, y;
        if (p < 4) {
            // FLOOR_MASK rows: (T,T),(T,F),(F,T),(F,F)
            x = (p & 2) ? ceilf(m0) : floorf(m0);
            y = (p & 1) ? ceilf(m1) : floorf(m1);
        } else if (p < 12) {
            int g = p - 4;
            const float* up = ug + (((((size_t)b * K_ + k) * CHUNK_ + c) * 8 + g) * 2);
            x = floorf(up[0] * (ONE_MEPS * RNGF));
            y = floorf(up[1] * (ONE_MEPS * RNGF));
        } else {
            int g = p - 12;
            const float* up = ur + (((((size_t)b * K_ + k) * CHUNK_ + c) * 8 + g) * 2);
            float mn0 = rintf(m0), mn1 = rintf(m1);   // round-half-even
            float lo0 = mn0 - 4.f; if (lo0 < 0.f) lo0 = 0.f;
            if (mn0 + 4.f > RNGF) lo0 = RNGF - 8.f;
            float lo1 = mn1 - 4.f; if (lo1 < 0.f) lo1 = 0.f;
            if (mn1 + 4.f > RNGF) lo1 = RNGF - 8.f;
            x = floorf(up[0] * (ONE_MEPS * 8.f) + lo0);
            y = floorf(up[1] * (ONE_MEPS * 8.f) + lo1);
        }
        int xi = (int)x, yi = (int)y;
        s_x[l] = xi;  s_y[l] = yi;
        unsigned long long hx = (unsigned long long)(xi + 1);
        unsigned long long hy = (unsigned long long)(yi + 1);
        s_h[l] = hx * hx * hy * hy * hy;             // (x+1)^2 (y+1)^3
    }
    __syncthreads();

    // ---- Phase 3: duplicate mask (stable-argsort semantics: first index wins) ----
    for (int l = tid; l < L_; l += 256) {
        unsigned long long h = s_h[l];
        float sc = 1.f;
        for (int j = 0; j < l; ++j)
            if (s_h[j] == h) { sc = 0.f; break; }
        s_ds[l] = sc;
    }
    if (tid < CHUNK_) s_denom[tid] = 0.f;
    for (int l = tid; l < L_; l += 256) s_val[l] = 0.f;
    __syncthreads();

    const int wave  = tid >> 5;
    const int lane  = tid & 31;
    const int row   = lane & 15;
    const int khalf = lane >> 4;       // A lanes 0-15: K=0,1 ; lanes 16-31: K=2,3
    const int kbase = khalf * 2;
    const int col   = row;             // B/D column within N-tile

    // ---- Pass A: WMMA quadratic form -> column denominators ----
    for (int t = wave * 5; t < wave * 5 + 5; ++t) {
        for (int n = 0; n < 2; ++n) {
            int la = t * 16 + row;
            float fx = (float)s_x[la];
            float fy = (float)s_y[la];
            v2f a, bb;
            a[0] = kbase ? fx : fx * fx;   // A row = [x^2, y^2, x, y]
            a[1] = kbase ? fy : fy * fy;
            int c = n * 16 + col;
            bb[0] = sB[kbase][c];
            bb[1] = sB[kbase + 1][c];
            v8f acc = {};
            v8f d = __builtin_amdgcn_wmma_f32_16x16x4_f32(
                false, a, false, bb, (short)0, acc, false, false);
            float t3c = s_t3[c];
            float colsum = 0.f;
#pragma unroll
            for (int v = 0; v < 8; ++v) {
                int l2 = t * 16 + v + khalf * 8;
                colsum += __expf(-0.5f * (d[v] + t3c)) * s_ds[l2];
            }
            atomicAdd(&s_denom[c], colsum);
        }
    }
    __syncthreads();
    if (tid < CHUNK_) s_vr[tid] = s_value[tid] / s_denom[tid];
    __syncthreads();

    // ---- Pass B: recompute WMMA, accumulate per-point values ----
    for (int t = wave * 5; t < wave * 5 + 5; ++t) {
        for (int n = 0; n < 2; ++n) {
            int la = t * 16 + row;
            float fx = (float)s_x[la];
            float fy = (float)s_y[la];
            v2f a, bb;
            a[0] = kbase ? fx : fx * fx;
            a[1] = kbase ? fy : fy * fy;
            int c = n * 16 + col;
            bb[0] = sB[kbase][c];
            bb[1] = sB[kbase + 1][c];
            v8f acc = {};
            v8f d = __builtin_amdgcn_wmma_f32_16x16x4_f32(
                false, a, false, bb, (short)0, acc, false, false);
            float t3c = s_t3[c];
            float vr  = s_vr[c];
#pragma unroll
            for (int v = 0; v < 8; ++v) {
                int l2 = t * 16 + v + khalf * 8;
                float contrib = __expf(-0.5f * (d[v] + t3c)) * s_ds[l2] * vr;
                // sum over the 16 columns held by this 16-lane half
                contrib += __shfl_xor(contrib, 1, 32);
                contrib += __shfl_xor(contrib, 2, 32);
                contrib += __shfl_xor(contrib, 4, 32);
                contrib += __shfl_xor(contrib, 8, 32);
                if (row == 0) atomicAdd(&s_val[l2], contrib);
            }
        }
    }
    __syncthreads();

    // ---- Scatter: vals * input[b, y] -> out[b, x] ----
    const float* inb  = input + (size_t)b * IN_;
    float*       outb = out   + (size_t)b * OUT_;
    for (int l = tid; l < L_; l += 256) {
        float g = s_val[l] * inb[s_y[l]];
        atomicAdd(&outb[s_x[l]], g);
    }
}

extern "C" void kernel_launch(void* const* d_in, const int* in_sizes, int n_in,
                              void* d_out, int out_size, void* d_ws, size_t ws_size,
                              hipStream_t stream) {
    const float* input = (const float*)d_in[0];
    const float* res   = (const float*)d_in[1];
    const float* bias  = (const float*)d_in[2];
    const float* ug    = (const float*)d_in[3];
    const float* ur    = (const float*)d_in[4];
    // d_in[5] (temp_indices) is fully overwritten by the scatter indices in the
    // reference (R==2 covers both columns) -> unused.
    float* out = (float*)d_out;

    init_out_kernel<<<(B_ * OUT_ + 255) / 256, 256, 0, stream>>>(bias, out);
    hyper_kernel<<<B_ * K_, 256, 0, stream>>>(input, res, ug, ur, out);
}